// CoreSnapshotEncoder_3092376453302
// MI455X (gfx1250) — compile-verified
//
#include <hip/hip_runtime.h>
#include <float.h>

// Problem constants (fixed by the reference): C=64 cores, D=128 emb dim.
#define C_CORES 64
#define DIM     128

typedef float v2f __attribute__((ext_vector_type(2)));
typedef float v4f __attribute__((ext_vector_type(4)));
typedef float v8f __attribute__((ext_vector_type(8)));

// ---------------------------------------------------------------------------
// Kernel 0: initialize workspace (segmax = -FLT_MAX, counts = 0) every call.
// ---------------------------------------------------------------------------
__global__ void init_ws_kernel(float* __restrict__ segmax, int* __restrict__ counts) {
    int i = blockIdx.x * blockDim.x + threadIdx.x;
    if (i < C_CORES * DIM) segmax[i] = -FLT_MAX;
    if (i < C_CORES)       counts[i] = 0;
}

// ---------------------------------------------------------------------------
// Kernel 1: fused segment_max + segment_count (the bandwidth-bound pass:
// 102.4 MB read-once -> ~4.4 us floor at 23.3 TB/s).
// One wave32 per row: 32 lanes x float4 = 128 floats = one full row, coalesced
// global_load_b128. Loads are NON-TEMPORAL: the stream is read exactly once,
// so don't let it evict useful lines from the 192 MB L2.
// Per-block privatized [64][128] max tile in LDS; peek-then-atomic keeps the
// LDS atomic unit idle once the block-local max saturates; merge with
// device-scope float atomic max (global_atomic_max_num_f32).
// ---------------------------------------------------------------------------
__global__ __launch_bounds__(256) void seg_max_kernel(
    const int*   __restrict__ assign,   // [Q]
    const float* __restrict__ qe,       // [Q][128]
    float*       __restrict__ gmax,     // [64][128] ws
    int*         __restrict__ gcnt,     // [64] ws
    int Q) {
    __shared__ float lmax[C_CORES * DIM];   // 32 KB
    __shared__ int   lcnt[C_CORES];

    const int tid = threadIdx.x;
    for (int i = tid; i < C_CORES * DIM; i += blockDim.x) lmax[i] = -FLT_MAX;
    if (tid < C_CORES) lcnt[tid] = 0;
    __syncthreads();

    const int lane   = tid & 31;
    const int wave   = (blockIdx.x * blockDim.x + tid) >> 5;
    const int nwaves = (gridDim.x * blockDim.x) >> 5;

    for (int r = wave; r < Q; r += nwaves) {
        const int core = __builtin_nontemporal_load(assign + r);   // wave-uniform
        const v4f v = __builtin_nontemporal_load(
            (const v4f*)(qe + (size_t)r * DIM + lane * 4));        // TH=NT stream
        float* base = &lmax[core * DIM + lane * 4];
        const v4f cur = *(const v4f*)base;                         // cheap LDS peek
        // Only touch the LDS atomic unit when we actually improve the max.
        if (v.x > cur.x) __hip_atomic_fetch_max(base + 0, v.x, __ATOMIC_RELAXED, __HIP_MEMORY_SCOPE_WORKGROUP);
        if (v.y > cur.y) __hip_atomic_fetch_max(base + 1, v.y, __ATOMIC_RELAXED, __HIP_MEMORY_SCOPE_WORKGROUP);
        if (v.z > cur.z) __hip_atomic_fetch_max(base + 2, v.z, __ATOMIC_RELAXED, __HIP_MEMORY_SCOPE_WORKGROUP);
        if (v.w > cur.w) __hip_atomic_fetch_max(base + 3, v.w, __ATOMIC_RELAXED, __HIP_MEMORY_SCOPE_WORKGROUP);
        if (lane == 0) atomicAdd(&lcnt[core], 1);
    }
    __syncthreads();

    // Merge block-private tile into the global accumulator.
    for (int i = tid; i < C_CORES * DIM; i += blockDim.x) {
        const float v = lmax[i];
        if (v > -FLT_MAX)
            __hip_atomic_fetch_max(&gmax[i], v, __ATOMIC_RELAXED, __HIP_MEMORY_SCOPE_AGENT);
    }
    if (tid < C_CORES && lcnt[tid] > 0) atomicAdd(&gcnt[tid], lcnt[tid]);
}

// ---------------------------------------------------------------------------
// Kernel 2: GNN layer on the 64x128 core embeddings, full-f32 WMMA
// (precision-exact vs the f32 reference; the GEMMs are only ~2.7 MFLOP so
// there is zero incentive to drop precision).
//   E   = where(cnt>0, segmax, padding)              [64][128] (LDS)
//   T   = E @ W_msg                                  [64][128] (LDS)
//   out = relu(E @ W_self + core_con @ T + b)        [64][128]
// 32 waves (1024 threads) = 4x8 grid of 16x16 tiles; V_WMMA_F32_16X16X4_F32.
//
// Fragment layouts (ISA 7.12.2, wave32):
//   A (16x4 f32):  M = lane&15; VGPR0 holds K = 2*(lane>>4), VGPR1 holds K+1.
//   B (4x16 f32):  N = lane&15; VGPR0 holds K = 2*(lane>>4), VGPR1 holds K+1.
//   C/D (16x16):   N = lane&15; VGPR v holds M = v + 8*(lane>>4).
// ---------------------------------------------------------------------------
__global__ __launch_bounds__(1024) void gnn_wmma_kernel(
    const float* __restrict__ gmax,      // [64][128] ws
    const int*   __restrict__ gcnt,      // [64] ws
    const float* __restrict__ padding,   // [128]
    const float* __restrict__ core_con,  // [64][64]
    const float* __restrict__ W_self,    // [128][128]
    const float* __restrict__ W_msg,     // [128][128]
    const float* __restrict__ bias,      // [128]
    float*       __restrict__ out) {     // [64][128]
    __shared__ float E[C_CORES * DIM];   // 32 KB
    __shared__ float T[C_CORES * DIM];   // 32 KB

    const int tid = threadIdx.x;

    // Stage 0: materialize core embeddings (padding for empty segments).
    for (int i = tid; i < C_CORES * DIM; i += blockDim.x) {
        const int c = i >> 7, d = i & (DIM - 1);
        E[i] = (gcnt[c] > 0) ? gmax[i] : padding[d];
    }
    __syncthreads();

    const int lane = tid & 31;
    const int w    = tid >> 5;      // wave id 0..31
    const int tm   = w >> 3;        // row-tile 0..3
    const int tn   = w & 7;         // col-tile 0..7
    const int hl   = lane >> 4;     // lane half
    const int l15  = lane & 15;

    const int arow = tm * 16 + l15; // A-fragment row (M)
    const int bcol = tn * 16 + l15; // B-fragment col (N)

    // Stage 1: T = E @ W_msg
    v8f acc = {};
    for (int kk = 0; kk < DIM; kk += 4) {
        const int ka = kk + 2 * hl;
        v2f a; a.x = E[arow * DIM + ka];        a.y = E[arow * DIM + ka + 1];
        v2f b; b.x = W_msg[ka * DIM + bcol];    b.y = W_msg[(ka + 1) * DIM + bcol];
        acc = __builtin_amdgcn_wmma_f32_16x16x4_f32(false, a, false, b,
                                                    (short)0, acc, false, false);
    }
    {
        const int row0 = tm * 16 + 8 * hl;
        const int col  = tn * 16 + l15;
        #pragma unroll
        for (int v = 0; v < 8; ++v) T[(row0 + v) * DIM + col] = acc[v];
    }
    __syncthreads();

    // Stage 2: acc = E @ W_self  (+)=  core_con @ T, then bias + relu.
    v8f acc2 = {};
    for (int kk = 0; kk < DIM; kk += 4) {
        const int ka = kk + 2 * hl;
        v2f a; a.x = E[arow * DIM + ka];        a.y = E[arow * DIM + ka + 1];
        v2f b; b.x = W_self[ka * DIM + bcol];   b.y = W_self[(ka + 1) * DIM + bcol];
        acc2 = __builtin_amdgcn_wmma_f32_16x16x4_f32(false, a, false, b,
                                                     (short)0, acc2, false, false);
    }
    for (int kk = 0; kk < C_CORES; kk += 4) {
        const int ka = kk + 2 * hl;
        v2f a; a.x = core_con[arow * C_CORES + ka]; a.y = core_con[arow * C_CORES + ka + 1];
        v2f b; b.x = T[ka * DIM + bcol];            b.y = T[(ka + 1) * DIM + bcol];
        acc2 = __builtin_amdgcn_wmma_f32_16x16x4_f32(false, a, false, b,
                                                     (short)0, acc2, false, false);
    }
    {
        const int row0 = tm * 16 + 8 * hl;
        const int col  = tn * 16 + l15;
        const float bv = bias[col];
        #pragma unroll
        for (int v = 0; v < 8; ++v) {
            const float r = acc2[v] + bv;
            out[(row0 + v) * DIM + col] = r > 0.f ? r : 0.f;
        }
    }
}

// ---------------------------------------------------------------------------
// Launcher
// ---------------------------------------------------------------------------
extern "C" void kernel_launch(void* const* d_in, const int* in_sizes, int n_in,
                              void* d_out, int out_size, void* d_ws, size_t ws_size,
                              hipStream_t stream) {
    const int*   prev_assign = (const int*)  d_in[0];
    const float* q_emb       = (const float*)d_in[1];
    const float* padding     = (const float*)d_in[2];
    const float* core_con    = (const float*)d_in[3];
    const float* W_self      = (const float*)d_in[4];
    const float* W_msg       = (const float*)d_in[5];
    const float* bias        = (const float*)d_in[6];
    float*       out         = (float*)d_out;
    const int Q = in_sizes[0];

    float* segmax = (float*)d_ws;                                        // 64*128 f32
    int*   counts = (int*)((char*)d_ws + C_CORES * DIM * sizeof(float)); // 64 i32

    init_ws_kernel<<<(C_CORES * DIM + 255) / 256, 256, 0, stream>>>(segmax, counts);
    seg_max_kernel<<<256, 256, 0, stream>>>(prev_assign, q_emb, segmax, counts, Q);
    gnn_wmma_kernel<<<1, 1024, 0, stream>>>(segmax, counts, padding, core_con,
                                            W_self, W_msg, bias, out);
}